// Chimera_80934363725826
// MI455X (gfx1250) — compile-verified
//
#include <hip/hip_runtime.h>

typedef __attribute__((ext_vector_type(2))) float v2f;
typedef __attribute__((ext_vector_type(4))) float v4f;
typedef __attribute__((ext_vector_type(8))) float v8f;

#define H_IMG 14
#define W_IMG 14
#define LNODES 196
#define NB 32
#define NHEADS 24
#define PDIM 64
#define SDT (NB * NHEADS * LNODES)   // stride between dt direction planes

// One workgroup per (b,h): solve (I-A) z = x by forward substitution over the
// 14 grid rows; each row's triangular left-scan is a padded 16x16 @ 16x64
// matmul done with V_WMMA_F32_16X16X4_F32 (4 waves x 4 k-steps).
// x is read exactly once (b128) and y written exactly once (b128, deferred by
// one row so the store uses the vectorized phase-1 layout) -> 77 MB HBM floor.
__launch_bounds__(128, 4)
__global__ void chimera_dag_mix(const float* __restrict__ dt,
                                const float* __restrict__ dt_bias,
                                const float* __restrict__ x,
                                const float* __restrict__ Dskip,
                                float* __restrict__ y)
{
    __shared__ float sAleft[LNODES];
    __shared__ float sAtop[LNODES];
    __shared__ float sT[16][17];                    // padded: no bank conflicts
    __shared__ alignas(16) float sZin[16][68];      // row-step RHS (rows 14,15 = 0)
    __shared__ alignas(16) float sZprev[16][68];    // previous grid-row solution

    const int bh   = blockIdx.x;          // 0..B*NH-1
    const int h    = bh % NHEADS;
    const int tid  = threadIdx.x;         // 0..127 (4 wave32s)
    const int lane = tid & 31;
    const int wave = tid >> 5;            // 0..3 -> 16-wide channel tile
    const int col0 = wave * 16;
    const int mrow = lane & 15;
    const int ksel = (lane >> 4) << 1;    // 0 for lanes 0-15, 2 for lanes 16-31

    const float bias = dt_bias[h];
    const float dsk  = Dskip[h];
    const float* __restrict__ xb = x + (size_t)bh * (LNODES * PDIM);
    float* __restrict__ yb       = y + (size_t)bh * (LNODES * PDIM);

    // ---- edge coefficients: a = 0.95 * sigmoid(-(dt+bias)) / sqrt(nie) ----
    for (int node = tid; node < LNODES; node += 128) {
        int row = node / W_IMG, col = node % W_IMG;
        float nie = 2.0f - (col == 0 ? 1.0f : 0.0f) - (row == 0 ? 1.0f : 0.0f);
        if (nie < 1e-6f) nie = 1.0f;                    // top-left corner
        float rn = 0.95f * rsqrtf(nie);
        float v0 = dt[0 * SDT + bh * LNODES + node] + bias;   // left->right edge
        float v1 = dt[1 * SDT + bh * LNODES + node] + bias;   // top->bottom edge
        sAleft[node] = (col > 0) ? rn / (1.0f + expf(v0)) : 0.0f;
        sAtop [node] = (row > 0) ? rn / (1.0f + expf(v1)) : 0.0f;
    }
    // one-time zero: z_prev (row 0 has a_top = 0, avoid 0*garbage NaN),
    // zin pad rows 14,15 (never touched again), T pad rows 14,15.
    {
        v4f zv = {};
        v4f* zp = (v4f*)&sZprev[0][0];
        for (int i = tid; i < (16 * 68) / 4; i += 128) zp[i] = zv;
        v4f* zi = (v4f*)&sZin[14][0];                 // rows 14,15 (row-aligned: 68=17*4)
        for (int i = tid; i < (2 * 68) / 4; i += 128) zi[i] = zv;
        if (tid < 32) sT[14 + (tid >> 4)][tid & 15] = 0.0f;
    }
    __syncthreads();

    // phase-1 ownership: thread -> (grid column c, 8 contiguous channels at p0)
    const int  zc  = tid >> 3;               // 0..15
    const int  zp0 = (tid & 7) * 8;          // 0,8,...,56
    const bool act = (zc < W_IMG);

    v4f xprev0 = {}, xprev1 = {};            // row r-1 x tile kept in registers

    for (int r = 0; r < H_IMG; ++r) {
        // ---- phase 1: y_{r-1} flush + RHS for this row (all b128) ----
        if (act) {
            int node = r * W_IMG + zc;
            float at = sAtop[node];
            const v4f* xp = (const v4f*)(xb + node * PDIM + zp0);
            v4f x0 = xp[0], x1 = xp[1];
            const v4f* pp = (const v4f*)&sZprev[zc][zp0];   // z of row r-1
            v4f p0 = pp[0], p1 = pp[1];
            if (r > 0) {                                    // y_{r-1} = z + D*x
                v4f* yp = (v4f*)(yb + (node - W_IMG) * PDIM + zp0);
                yp[0] = p0 + dsk * xprev0;
                yp[1] = p1 + dsk * xprev1;
            }
            v4f* op = (v4f*)&sZin[zc][zp0];                 // zin = x + a_top*z
            op[0] = x0 + at * p0;
            op[1] = x1 + at * p1;
            xprev0 = x0; xprev1 = x1;
        }
        // ---- build T_r: T[c][c'] = a_left[c] * T[c-1][c'], T[c][c] = 1 ----
        if (tid < 16) {
            float t = 0.0f;                       // lane 'tid' owns column c'
            for (int c = 0; c < W_IMG; ++c) {
                float al = sAleft[r * W_IMG + c];
                t = (tid == c) ? 1.0f : al * t;   // stays 0 above the diagonal
                sT[c][tid] = t;
            }
        }
        __syncthreads();

        // ---- phase 2: z_row = T_r @ zin  (16x16 @ 16x64, f32 WMMA) ----
        v8f acc = {};
#pragma unroll
        for (int k0 = 0; k0 < 16; k0 += 4) {
            v2f a, b;
            // A (16x4): VGPR0 = K k0/k0+2 per lane half, VGPR1 = K k0+1/k0+3
            a.x = sT[mrow][k0 + ksel];
            a.y = sT[mrow][k0 + ksel + 1];
            // B (4x16): same K striping, N = lane&15 within the wave's tile
            b.x = sZin[k0 + ksel][col0 + mrow];
            b.y = sZin[k0 + ksel + 1][col0 + mrow];
            acc = __builtin_amdgcn_wmma_f32_16x16x4_f32(
                false, a, false, b, (short)0, acc, false, false);
        }

        // ---- spill accumulator to LDS only (C/D: VGPR i -> M=i, +8 upper) ----
#pragma unroll
        for (int i = 0; i < 8; ++i) {
            int M = i + ((lane >> 4) << 3);
            if (M < W_IMG) sZprev[M][col0 + mrow] = acc[i];  // feeds next row
        }
        __syncthreads();
    }

    // ---- flush last row's y ----
    if (act) {
        const v4f* pp = (const v4f*)&sZprev[zc][zp0];
        v4f* yp = (v4f*)(yb + ((H_IMG - 1) * W_IMG + zc) * PDIM + zp0);
        yp[0] = pp[0] + dsk * xprev0;
        yp[1] = pp[1] + dsk * xprev1;
    }
}

extern "C" void kernel_launch(void* const* d_in, const int* in_sizes, int n_in,
                              void* d_out, int out_size, void* d_ws, size_t ws_size,
                              hipStream_t stream) {
    (void)in_sizes; (void)n_in; (void)d_ws; (void)ws_size; (void)out_size;
    const float* dt      = (const float*)d_in[0];
    const float* dt_bias = (const float*)d_in[1];
    const float* x       = (const float*)d_in[2];
    const float* Dskip   = (const float*)d_in[3];
    float* y             = (float*)d_out;
    chimera_dag_mix<<<NB * NHEADS, 128, 0, stream>>>(dt, dt_bias, x, Dskip, y);
}